// HideAdapter_1073741824233
// MI455X (gfx1250) — compile-verified
//
#include <hip/hip_runtime.h>

typedef float v2f __attribute__((ext_vector_type(2)));
typedef float v8f __attribute__((ext_vector_type(8)));
typedef int   v4i __attribute__((ext_vector_type(4)));

#define BATCH 128
#define SEQ   197
#define DIM   768
#define RANK  64
#define NL    12          // depth levels (L)
#define TS    32          // S-rows per workgroup
#define NT    ((SEQ + TS - 1) / TS)   // 7 s-tiles
#define XSTR  772         // padded LDS row stride for x tile (bank-conflict-free b64)
#define DSTR  66          // padded LDS row stride for down tile

#if defined(__AMDGCN__) && __has_builtin(__builtin_amdgcn_global_load_async_to_lds_b128) && __has_builtin(__builtin_amdgcn_s_wait_asynccnt)
#define USE_ASYNC_LDS 1
typedef __attribute__((address_space(1))) v4i* gvec_t;   // global (AS1) int4*
typedef __attribute__((address_space(3))) v4i* lvec_t;   // LDS    (AS3) int4*
#else
#define USE_ASYNC_LDS 0
#endif

__global__ __launch_bounds__(256)
void hide_adapter_kernel(const float* __restrict__ x,
                         const float* __restrict__ down_proj,   // [P][L][D][R]
                         const float* __restrict__ down_bias,   // [P][L][R]
                         const float* __restrict__ up_proj,     // [P][L][R][D]
                         const float* __restrict__ up_bias,     // [P][L][D]
                         const int*   __restrict__ task_id,
                         const int*   __restrict__ depth_id,
                         float* __restrict__ out)
{
    __shared__ float x_lds[TS * XSTR];     // ~96.5 KB
    __shared__ float down_lds[TS * DSTR];  // ~8.25 KB

    const int wg = blockIdx.x;
    const int b  = wg / NT;
    const int st = wg % NT;
    const int s0 = st * TS;

    const int tid  = threadIdx.x;
    const int lane = tid & 31;
    const int wave = tid >> 5;            // 0..7

    const int  tk = task_id[b];
    const int  dl = depth_id[b];
    const size_t adapter = (size_t)tk * NL + dl;
    const float* __restrict__ dp = down_proj + adapter * (size_t)DIM * RANK;
    const float* __restrict__ db = down_bias + adapter * RANK;
    const float* __restrict__ uw = up_proj   + adapter * (size_t)RANK * DIM;
    const float* __restrict__ ub = up_bias   + adapter * DIM;
    const float* __restrict__ xb = x   + (size_t)b * SEQ * DIM;
    float*       __restrict__ ob = out + (size_t)b * SEQ * DIM;

    // ---- warm W_down into the WGP while we stage x (192 KB, L2-resident) ----
    #pragma unroll
    for (int i = 0; i < (DIM * RANK) / (256 * 32); ++i) {
        __builtin_prefetch(dp + (i * 256 + tid) * 32, 0, 3);   // global_prefetch_b8
    }

    // ---- stage x tile into LDS (clamp rows past SEQ; they are never stored) ----
#if USE_ASYNC_LDS
    // Direct global->LDS async copy (no VGPR round-trip), tracked by ASYNCcnt.
    for (int i = tid; i < TS * (DIM / 4); i += 256) {
        const int row  = i / (DIM / 4);
        const int col4 = i % (DIM / 4);
        int srow = s0 + row;
        if (srow >= SEQ) srow = SEQ - 1;
        gvec_t g = (gvec_t)(xb + (size_t)srow * DIM + col4 * 4);
        lvec_t l = (lvec_t)(x_lds + row * XSTR + col4 * 4);
        __builtin_amdgcn_global_load_async_to_lds_b128(g, l, 0, 0);
    }
    __builtin_amdgcn_s_wait_asynccnt(0);
#else
    for (int i = tid; i < TS * (DIM / 4); i += 256) {
        const int row  = i / (DIM / 4);
        const int col4 = i % (DIM / 4);
        int srow = s0 + row;
        if (srow >= SEQ) srow = SEQ - 1;
        const float4 v = ((const float4*)(xb + (size_t)srow * DIM))[col4];
        ((float4*)(x_lds + row * XSTR))[col4] = v;
    }
#endif
    __syncthreads();

    const int col   = lane & 15;   // A row within tile / C-D column
    const int khalf = lane >> 4;   // K sub-slot select (0 or 1)

    // ---- Phase A: down = relu(x_tile @ W_down + db)  -> down_lds [TS][RANK] ----
    {
        const int mt = wave >> 2;          // 0..1  (M tile of 16)
        const int nt = wave & 3;           // 0..3  (N tile of 16)
        const float* aptr = &x_lds[(mt * 16 + col) * XSTR + khalf * 2];
        const float* bptr = dp + (size_t)(khalf * 2) * RANK + nt * 16 + col;
        v8f acc = {};
        #pragma unroll 8
        for (int k = 0; k < DIM; k += 4) {
            v2f a = *(const v2f*)aptr;                    // ds_load_b64 (2addr-merged)
            v2f bv; bv.x = bptr[0]; bv.y = bptr[RANK];    // coalesced b32 x2 (L2)
            acc = __builtin_amdgcn_wmma_f32_16x16x4_f32(
                      false, a, false, bv, (short)0, acc, false, false);
            aptr += 4;
            bptr += 4 * RANK;
        }
        const int   n    = nt * 16 + col;
        const float bias = db[n];
        #pragma unroll
        for (int j = 0; j < 8; ++j) {
            const int r = mt * 16 + j + 8 * khalf;        // C/D row layout
            float v = acc[j] + bias;
            down_lds[r * DSTR + n] = v > 0.0f ? v : 0.0f;
        }
    }

    // ---- warm W_up into the WGP before phase B consumes it ----
    #pragma unroll
    for (int i = 0; i < (RANK * DIM) / (256 * 32); ++i) {
        __builtin_prefetch(uw + (i * 256 + tid) * 32, 0, 3);
    }
    __syncthreads();

    // ---- Phase B: out = down @ W_up + ub + x  (SCALE == 1.0) ----
    for (int tile = wave; tile < 2 * (DIM / 16); tile += 8) {
        const int mt = tile / (DIM / 16);
        const int nt = tile % (DIM / 16);
        const float* aptr = &down_lds[(mt * 16 + col) * DSTR + khalf * 2];
        const float* bptr = uw + (size_t)(khalf * 2) * DIM + nt * 16 + col;
        v8f acc = {};
        #pragma unroll
        for (int k = 0; k < RANK; k += 4) {
            v2f a = *(const v2f*)aptr;
            v2f bv; bv.x = bptr[0]; bv.y = bptr[DIM];
            acc = __builtin_amdgcn_wmma_f32_16x16x4_f32(
                      false, a, false, bv, (short)0, acc, false, false);
            aptr += 4;
            bptr += 4 * DIM;
        }
        const int   d    = nt * 16 + col;
        const float bias = ub[d];
        #pragma unroll
        for (int j = 0; j < 8; ++j) {
            const int r = mt * 16 + j + 8 * khalf;
            const int s = s0 + r;
            if (s < SEQ) {
                ob[(size_t)s * DIM + d] = acc[j] + bias + x_lds[r * XSTR + d];
            }
        }
    }
}

extern "C" void kernel_launch(void* const* d_in, const int* in_sizes, int n_in,
                              void* d_out, int out_size, void* d_ws, size_t ws_size,
                              hipStream_t stream) {
    (void)in_sizes; (void)n_in; (void)out_size; (void)d_ws; (void)ws_size;
    const float* x   = (const float*)d_in[0];
    const float* dpw = (const float*)d_in[1];
    const float* dbv = (const float*)d_in[2];
    const float* upw = (const float*)d_in[3];
    const float* ubv = (const float*)d_in[4];
    const int*   tk  = (const int*)d_in[5];
    const int*   dl  = (const int*)d_in[6];
    float* out = (float*)d_out;

    hide_adapter_kernel<<<dim3(BATCH * NT), dim3(256), 0, stream>>>(
        x, dpw, dbv, upw, ubv, tk, dl, out);
}